// GNN_31903017075422
// MI455X (gfx1250) — compile-verified
//
#include <hip/hip_runtime.h>

// ---------------------------------------------------------------------------
// GraphSAGE (3 layers) + MLP head for MI455X (gfx1250, wave32, WMMA).
//
//  * CSR built ONCE (degree count -> 1-block scan -> atomic fill), reused by
//    all 3 layers: avoids ~307M global float atomics of naive segment_sum.
//  * Aggregation: one wave per node, lane owns a float4 slice of the 128-wide
//    row; 25.6MB feature table is L2-resident (192MB L2). global_prefetch_b8
//    issued 2 edges ahead.
//  * GEMM: v_wmma_f32_16x16x4_f32 (exact fp32). Weights pre-swizzled into
//    B-fragment order in LDS -> one conflict-free ds_load_b64 per WMMA with
//    immediate offsets. Templated on NOUT/HAS2/RELU so the 32-step K loop is
//    fully unrolled with immediate addressing.
// ---------------------------------------------------------------------------

typedef float v2f __attribute__((ext_vector_type(2)));
typedef float v8f __attribute__((ext_vector_type(8)));

#define NN 50000
#define NE 800000
#define DH 128
#define DO 64

// ---------------------------- CSR construction ----------------------------

__global__ void k_zero_i32(int* __restrict__ p, int n) {
  int i = blockIdx.x * blockDim.x + threadIdx.x;
  if (i < n) p[i] = 0;
}

__global__ void k_count(const int* __restrict__ dst, int* __restrict__ cnt, int e) {
  int i = blockIdx.x * blockDim.x + threadIdx.x;
  if (i < e) atomicAdd(&cnt[dst[i]], 1);
}

// Single-block exclusive scan (1024 threads, chunked Hillis-Steele).
__global__ void k_scan(const int* __restrict__ cnt, int* __restrict__ row_ptr, int n) {
  __shared__ int sm[1024];
  __shared__ int base_s;
  if (threadIdx.x == 0) { base_s = 0; row_ptr[0] = 0; }
  __syncthreads();
  for (int start = 0; start < n; start += 1024) {
    int i = start + (int)threadIdx.x;
    sm[threadIdx.x] = (i < n) ? cnt[i] : 0;
    __syncthreads();
    for (int off = 1; off < 1024; off <<= 1) {
      int t = (threadIdx.x >= (unsigned)off) ? sm[threadIdx.x - off] : 0;
      __syncthreads();
      sm[threadIdx.x] += t;
      __syncthreads();
    }
    if (i < n) row_ptr[i + 1] = base_s + sm[threadIdx.x];
    __syncthreads();
    if (threadIdx.x == 1023) base_s += sm[1023];
    __syncthreads();
  }
}

__global__ void k_copy_i32(const int* __restrict__ a, int* __restrict__ b, int n) {
  int i = blockIdx.x * blockDim.x + threadIdx.x;
  if (i < n) b[i] = a[i];
}

__global__ void k_fill(const int* __restrict__ src, const int* __restrict__ dst,
                       int* __restrict__ cursor, int* __restrict__ col, int e) {
  int i = blockIdx.x * blockDim.x + threadIdx.x;
  if (i < e) {
    int p = atomicAdd(&cursor[dst[i]], 1);
    col[p] = src[i];
  }
}

// ------------------------- mean neighbor aggregation -----------------------
// One wave (32 lanes) per node; lane owns 4 consecutive floats of the row.
__global__ __launch_bounds__(256)
void k_aggregate(const float* __restrict__ h, const int* __restrict__ row_ptr,
                 const int* __restrict__ col, float* __restrict__ agg, int n) {
  int wave = (int)((blockIdx.x * blockDim.x + threadIdx.x) >> 5);
  int lane = (int)(threadIdx.x & 31);
  if (wave >= n) return;
  int beg = row_ptr[wave];
  int end = row_ptr[wave + 1];
  float ax = 0.f, ay = 0.f, az = 0.f, aw = 0.f;
  for (int e = beg; e < end; ++e) {
    int s = col[e];  // wave-uniform -> scalar load
    if (e + 2 < end) {
      int sp = col[e + 2];
      __builtin_prefetch(h + (size_t)sp * DH + lane * 4, 0, 3);  // global_prefetch_b8
    }
    const float4 v = *(const float4*)(h + (size_t)s * DH + lane * 4);
    ax += v.x; ay += v.y; az += v.z; aw += v.w;
  }
  float inv = (end > beg) ? 1.0f / (float)(end - beg) : 0.0f;
  float4 o; o.x = ax * inv; o.y = ay * inv; o.z = az * inv; o.w = aw * inv;
  *(float4*)(agg + (size_t)wave * DH + lane * 4) = o;
}

// ------------------------------- WMMA GEMM ---------------------------------
// out[M,NOUT] = act( A1[M,128] @ Wa + bias (+ A2[M,128] @ Wb) )
// One wave per 16x16 output tile; block = 256 threads = 8 waves = one M-tile
// across NOUT/16 N-tiles. M is a multiple of 16 (50000 = 3125*16).
//
// Weights are staged into LDS in B-FRAGMENT order: bsm[ntile][kstep][lane] is
// the float2 {W[k0][col], W[k0+1][col]} that `lane` feeds to WMMA at `kstep`
// (k0 = 4*kstep + 2*(lane>=16), col = 16*ntile + (lane&15)). A wave's read at
// each kstep touches 32 consecutive float2 = all 64 LDS banks: conflict-free
// ds_load_b64 with a 256B immediate-offset stride.
template <int NOUT, bool HAS2, bool RELU>
__global__ __launch_bounds__(256)
void k_gemm(const float* __restrict__ A1, const float* __restrict__ Wa,
            const float* __restrict__ bias,
            const float* __restrict__ A2, const float* __restrict__ Wb,
            float* __restrict__ out) {
  constexpr int K  = 128;
  constexpr int NT = NOUT / 16;   // N-tiles
  constexpr int KS = K / 4;       // 32 WMMA k-steps
  __shared__ float2 bsm[NT * KS * 32];  // <= 64 KB (NT=8)

  const int tid  = (int)threadIdx.x;
  const int wid  = tid >> 5;
  const int lane = tid & 31;
  const int half = lane >> 4;     // 0: lanes 0-15, 1: lanes 16-31
  const int r    = lane & 15;
  const int mtile = (int)blockIdx.x;
  const bool active = (wid < NT);
  const int col = (wid << 4) + r;
  const int row = (mtile << 4) + r;

  // ---- stage Wa into fragment order ----
  for (int j = tid; j < NT * KS * 32; j += 256) {
    const int lj = j & 31;
    const int ks = (j >> 5) & 31;
    const int nt = j >> 10;
    const int k0 = ks * 4 + ((lj >> 4) << 1);
    const int cj = nt * 16 + (lj & 15);
    float2 v; v.x = Wa[k0 * NOUT + cj]; v.y = Wa[(k0 + 1) * NOUT + cj];
    bsm[j] = v;
  }
  __syncthreads();

  v8f c0 = {}, c1 = {};  // two accumulator chains for ILP

  if (active) {
    const float*  a1 = A1 + (size_t)row * K + 2 * half;
    const float2* bp = bsm + wid * (KS * 32) + lane;
    #pragma unroll
    for (int ks = 0; ks < KS; ks += 2) {
      v2f a  = *(const v2f*)(a1 + ks * 4);
      v2f b  = *(const v2f*)(bp + ks * 32);
      c0 = __builtin_amdgcn_wmma_f32_16x16x4_f32(false, a, false, b, (short)0, c0,
                                                 false, false);
      v2f a2 = *(const v2f*)(a1 + ks * 4 + 4);
      v2f b2 = *(const v2f*)(bp + (ks + 1) * 32);
      c1 = __builtin_amdgcn_wmma_f32_16x16x4_f32(false, a2, false, b2, (short)0, c1,
                                                 false, false);
    }
  }

  if (HAS2) {
    __syncthreads();
    for (int j = tid; j < NT * KS * 32; j += 256) {
      const int lj = j & 31;
      const int ks = (j >> 5) & 31;
      const int nt = j >> 10;
      const int k0 = ks * 4 + ((lj >> 4) << 1);
      const int cj = nt * 16 + (lj & 15);
      float2 v; v.x = Wb[k0 * NOUT + cj]; v.y = Wb[(k0 + 1) * NOUT + cj];
      bsm[j] = v;
    }
    __syncthreads();
    if (active) {
      const float*  a2p = A2 + (size_t)row * K + 2 * half;
      const float2* bp  = bsm + wid * (KS * 32) + lane;
      #pragma unroll
      for (int ks = 0; ks < KS; ks += 2) {
        v2f a  = *(const v2f*)(a2p + ks * 4);
        v2f b  = *(const v2f*)(bp + ks * 32);
        c0 = __builtin_amdgcn_wmma_f32_16x16x4_f32(false, a, false, b, (short)0, c0,
                                                   false, false);
        v2f a2 = *(const v2f*)(a2p + ks * 4 + 4);
        v2f b2 = *(const v2f*)(bp + (ks + 1) * 32);
        c1 = __builtin_amdgcn_wmma_f32_16x16x4_f32(false, a2, false, b2, (short)0, c1,
                                                   false, false);
      }
    }
  }

  if (active) {
    const float bv = bias[col];
    // C/D layout: VGPR i holds M=i (lanes 0-15) / M=8+i (lanes 16-31).
    #pragma unroll
    for (int i = 0; i < 8; ++i) {
      float val = c0[i] + c1[i] + bv;
      if (RELU) val = fmaxf(val, 0.0f);
      out[(size_t)((mtile << 4) + 8 * half + i) * NOUT + col] = val;
    }
  }
}

// ------------------------------- launcher ----------------------------------

extern "C" void kernel_launch(void* const* d_in, const int* in_sizes, int n_in,
                              void* d_out, int out_size, void* d_ws, size_t ws_size,
                              hipStream_t stream) {
  const float* x   = (const float*)d_in[0];
  const int*   ei  = (const int*)d_in[1];   // [2, E]: row 0 = src, row 1 = dst
  const float* Wl0 = (const float*)d_in[2];
  const float* bl0 = (const float*)d_in[3];
  const float* Wr0 = (const float*)d_in[4];
  const float* Wl1 = (const float*)d_in[5];
  const float* bl1 = (const float*)d_in[6];
  const float* Wr1 = (const float*)d_in[7];
  const float* Wl2 = (const float*)d_in[8];
  const float* bl2 = (const float*)d_in[9];
  const float* Wr2 = (const float*)d_in[10];
  const float* W1  = (const float*)d_in[11];
  const float* b1  = (const float*)d_in[12];
  const float* W2  = (const float*)d_in[13];
  const float* b2  = (const float*)d_in[14];
  float* out = (float*)d_out;

  const int N = NN, E = NE;
  const int* src = ei;
  const int* dst = ei + E;

  // ---- workspace carve-up (~81 MB) ----
  char* ws = (char*)d_ws;
  float* agg = (float*)ws;                     ws += (size_t)N * DH * sizeof(float);
  float* hA  = (float*)ws;                     ws += (size_t)N * DH * sizeof(float);
  float* hB  = (float*)ws;                     ws += (size_t)N * DH * sizeof(float);
  int* row_ptr = (int*)ws;                     ws += (size_t)(N + 16) * sizeof(int);
  int* cursor  = (int*)ws;                     ws += (size_t)N * sizeof(int);
  int* counts  = (int*)ws;                     ws += (size_t)N * sizeof(int);
  int* col     = (int*)ws;                     ws += (size_t)E * sizeof(int);

  const int TB = 256;
  const int gN = (N + TB - 1) / TB;
  const int gE = (E + TB - 1) / TB;
  const int gAgg = (N * 32 + TB - 1) / TB;  // one wave per node
  const int gTiles = N / 16;                // 3125, exact

  // ---- build CSR once (reused by all 3 layers) ----
  k_zero_i32<<<gN, TB, 0, stream>>>(counts, N);
  k_count<<<gE, TB, 0, stream>>>(dst, counts, E);
  k_scan<<<1, 1024, 0, stream>>>(counts, row_ptr, N);
  k_copy_i32<<<gN, TB, 0, stream>>>(row_ptr, cursor, N);
  k_fill<<<gE, TB, 0, stream>>>(src, dst, cursor, col, E);

  // ---- layer 0: x -> hA ----
  k_aggregate<<<gAgg, TB, 0, stream>>>(x, row_ptr, col, agg, N);
  k_gemm<DH, true, true><<<gTiles, TB, 0, stream>>>(agg, Wl0, bl0, x, Wr0, hA);
  // ---- layer 1: hA -> hB ----
  k_aggregate<<<gAgg, TB, 0, stream>>>(hA, row_ptr, col, agg, N);
  k_gemm<DH, true, true><<<gTiles, TB, 0, stream>>>(agg, Wl1, bl1, hA, Wr1, hB);
  // ---- layer 2: hB -> hA ----
  k_aggregate<<<gAgg, TB, 0, stream>>>(hB, row_ptr, col, agg, N);
  k_gemm<DH, true, true><<<gTiles, TB, 0, stream>>>(agg, Wl2, bl2, hB, Wr2, hA);

  // ---- head: hA @ W1 + b1 -> agg (reuse); agg @ W2 + b2 -> out ----
  k_gemm<DH, false, false><<<gTiles, TB, 0, stream>>>(hA, W1, b1, nullptr, nullptr, agg);
  k_gemm<DO, false, false><<<gTiles, TB, 0, stream>>>(agg, W2, b2, nullptr, nullptr, out);

  (void)in_sizes; (void)n_in; (void)out_size; (void)ws_size;
}